// S2VDQN_12867722019085
// MI455X (gfx1250) — compile-verified
//
#include <hip/hip_runtime.h>

#define N_NODES 50000
#define N_EDGES 1600000
#define GRAPHS  16
#define HDIM    128
#define HOPS    3
#define NTILES  (N_NODES / 16)   // 3125 exact

typedef __attribute__((ext_vector_type(2))) float v2f;
typedef __attribute__((ext_vector_type(4))) float v4f;
typedef __attribute__((ext_vector_type(8))) float v8f;
typedef __attribute__((ext_vector_type(2))) int   v2i;

#if defined(__gfx1250__) && defined(__has_builtin)
#if __has_builtin(__builtin_amdgcn_global_load_async_to_lds_b64)
#define USE_ASYNC_LDS 1
#endif
#endif

// Stage B matrix (128x128 f32, row-major, B[k][n] = W[n][k]) into LDS in
// per-lane WMMA fragment order: slot (kk, j, lane) = { W[j*16+lane16][kk*4+2*hi],
// W[...][kk*4+2*hi+1] }.  8192 slots * 8B = 64KB, conflict-free ds_load_b64 reads.
__device__ __forceinline__ void stage_B(const float* __restrict__ W, v2f* ldsB) {
  for (int s = threadIdx.x; s < 8192; s += 256) {
    int lane = s & 31;
    int j    = (s >> 5) & 7;
    int kk   = s >> 8;
    int col  = j * 16 + (lane & 15);
    int kp   = kk * 4 + 2 * (lane >> 4);
    const float* src = W + col * HDIM + kp;
#ifdef USE_ASYNC_LDS
    __builtin_amdgcn_global_load_async_to_lds_b64(
        (__attribute__((address_space(1))) v2i*)src,
        (__attribute__((address_space(3))) v2i*)&ldsB[s],
        0, 0);
#else
    ldsB[s] = *(const v2f*)src;
#endif
  }
#ifdef USE_ASYNC_LDS
  asm volatile("s_wait_asynccnt 0" ::: "memory");
#endif
  __syncthreads();
}

// ---------------- utility ----------------
__global__ void zero_kernel(float* p, int n) {
  int i = blockIdx.x * blockDim.x + threadIdx.x;
  if (i < n) p[i] = 0.f;
}

// per-edge scalar scatter: p[n] = sum max(ea,0), m[n] = sum max(-ea,0) over row==n
__global__ void edge_scalar_kernel(const int* __restrict__ ei, const float* __restrict__ ea,
                                   float* __restrict__ p, float* __restrict__ m) {
  int e = blockIdx.x * blockDim.x + threadIdx.x;
  if (e >= N_EDGES) return;
  int r = ei[e];            // row = edge_index[0][e]
  float a = ea[e];
  atomicAdd(p + r, fmaxf(a, 0.f));
  atomicAdd(m + r, fmaxf(-a, 0.f));
}

// u3p[l][h] = sum_k W3[l][h][k]*relu(W4[l][k]);  u3m with relu(-W4)
__global__ void precompute_u_kernel(const float* __restrict__ W3, const float* __restrict__ W4,
                                    float* __restrict__ u3p, float* __restrict__ u3m) {
  int t = blockIdx.x * blockDim.x + threadIdx.x;
  if (t >= HOPS * HDIM) return;
  int l = t / HDIM, h = t % HDIM;
  const float* w3row = W3 + (l * HDIM + h) * HDIM;
  const float* w4    = W4 + l * HDIM;
  float sp = 0.f, sm = 0.f;
  for (int k = 0; k < HDIM; k++) {
    float w = w4[k];
    sp += w3row[k] * fmaxf(w, 0.f);
    sm += w3row[k] * fmaxf(-w, 0.f);
  }
  u3p[t] = sp; u3m[t] = sm;
}

// hop 0: node_emb = relu(x*W1[0] + p*u3p[0] + m*u3m[0])   (ne_aggr == 0)
__global__ void hop0_kernel(const float* __restrict__ x, const float* __restrict__ W1l,
                            const float* __restrict__ p, const float* __restrict__ m,
                            const float* __restrict__ u3pl, const float* __restrict__ u3ml,
                            float* __restrict__ emb) {
  int i = blockIdx.x * blockDim.x + threadIdx.x;
  if (i >= N_NODES * HDIM) return;
  int n = i >> 7, h = i & 127;
  float v = x[n] * W1l[h] + p[n] * u3pl[h] + m[n] * u3ml[h];
  emb[i] = fmaxf(v, 0.f);
}

// one wave32 per edge: aggr[row] += emb[col]   (128 floats, float4 per lane)
__global__ void scatter_kernel(const int* __restrict__ ei, const float* __restrict__ emb,
                               float* __restrict__ aggr) {
  int t = blockIdx.x * blockDim.x + threadIdx.x;
  int e = t >> 5;
  if (e >= N_EDGES) return;
  int lane = t & 31;
  int r = ei[e];
  int c = ei[N_EDGES + e];
  v4f v = *(const v4f*)(emb + c * HDIM + lane * 4);
  float* dst = aggr + r * HDIM + lane * 4;
  atomicAdd(dst + 0, v.x);
  atomicAdd(dst + 1, v.y);
  atomicAdd(dst + 2, v.z);
  atomicAdd(dst + 3, v.w);
}

// ------------- WMMA f32 GEMM for hops 1..2 -------------
// out_emb[n,h] = relu( sum_k aggr[n,k]*W2l[h,k] + x[n]*W1l[h] + p[n]*u3pl[h] + m[n]*u3ml[h] )
// One wave computes a 16-row x 128-col stripe: 8 accumulator tiles, K=128 in steps of 4.
__global__ __launch_bounds__(256) void gemm_hop_kernel(
    const float* __restrict__ aggr, const float* __restrict__ W2l,
    const float* __restrict__ x, const float* __restrict__ W1l,
    const float* __restrict__ p, const float* __restrict__ m,
    const float* __restrict__ u3pl, const float* __restrict__ u3ml,
    float* __restrict__ out_emb) {
  __shared__ v2f ldsB[8192];           // 64KB: B fragments, pre-swizzled
  stage_B(W2l, ldsB);                  // all waves participate (barrier inside)

  const int lane = threadIdx.x & 31;
  const int wave = threadIdx.x >> 5;
  const int tile = blockIdx.x * 8 + wave;
  if (tile >= NTILES) return;          // wave-uniform; EXEC stays all-ones
  const int lane16 = lane & 15;
  const int hi = lane >> 4;
  const int row_base = tile * 16;

  v8f acc[8];
  #pragma unroll
  for (int j = 0; j < 8; j++)
    #pragma unroll
    for (int r = 0; r < 8; r++) acc[j][r] = 0.f;

  // A fragment source: lane<16 reads A[m][k..k+1], lane>=16 reads A[m][k+2..k+3]
  const float* aPtr = aggr + (row_base + lane16) * HDIM + 2 * hi;
  const v2f* bLane = ldsB + lane;

  #pragma unroll 4
  for (int kk = 0; kk < 32; kk++) {
    v2f a = *(const v2f*)(aPtr + 4 * kk);
    #pragma unroll
    for (int j = 0; j < 8; j++) {
      v2f b = bLane[(kk * 8 + j) * 32];
      acc[j] = __builtin_amdgcn_wmma_f32_16x16x4_f32(
          false, a, false, b, (short)0, acc[j], false, false);
    }
  }

  // fused bias + relu epilogue
  float w1v[8], upv[8], umv[8];
  #pragma unroll
  for (int j = 0; j < 8; j++) {
    int h = j * 16 + lane16;
    w1v[j] = W1l[h]; upv[j] = u3pl[h]; umv[j] = u3ml[h];
  }
  #pragma unroll
  for (int r = 0; r < 8; r++) {
    int row = row_base + r + 8 * hi;
    float xv = x[row], pv = p[row], mv = m[row];
    #pragma unroll
    for (int j = 0; j < 8; j++) {
      int h = j * 16 + lane16;
      float v = acc[j][r] + xv * w1v[j] + pv * upv[j] + mv * umv[j];
      out_emb[row * HDIM + h] = fmaxf(v, 0.f);
    }
  }
}

// pooled[g][h] += sum over nodes of graph g (LDS partial, then global atomic)
__global__ void pool_kernel(const float* __restrict__ emb, const int* __restrict__ batch,
                            float* __restrict__ pooled) {
  __shared__ float acc[GRAPHS * HDIM];
  for (int i = threadIdx.x; i < GRAPHS * HDIM; i += blockDim.x) acc[i] = 0.f;
  __syncthreads();
  const int total = N_NODES * HDIM;
  int stride = gridDim.x * blockDim.x;
  for (int i = blockIdx.x * blockDim.x + threadIdx.x; i < total; i += stride) {
    int n = i >> 7, h = i & 127;
    atomicAdd(&acc[batch[n] * HDIM + h], emb[i]);
  }
  __syncthreads();
  for (int i = threadIdx.x; i < GRAPHS * HDIM; i += blockDim.x)
    atomicAdd(&pooled[i], acc[i]);
}

// sum_emb[g][h] = sum_k pooled[g][k] * W6[h][k]
__global__ void sumemb_kernel(const float* __restrict__ pooled, const float* __restrict__ W6,
                              float* __restrict__ sum_emb) {
  int t = blockIdx.x * blockDim.x + threadIdx.x;
  if (t >= GRAPHS * HDIM) return;
  int g = t >> 7, h = t & 127;
  const float* pr = pooled + g * HDIM;
  const float* wr = W6 + h * HDIM;
  float s = 0.f;
  for (int k = 0; k < HDIM; k++) s += pr[k] * wr[k];
  sum_emb[t] = s;
}

// tg[g] = sum_h W5[0][h] * relu(sum_emb[g][h])
__global__ void tg_kernel(const float* __restrict__ sum_emb, const float* __restrict__ W5,
                          float* __restrict__ tg) {
  int g = blockIdx.x;
  int h = threadIdx.x;   // 128 threads = 4 waves
  float v = W5[h] * fmaxf(sum_emb[g * HDIM + h], 0.f);
  v += __shfl_xor(v, 1, 32);
  v += __shfl_xor(v, 2, 32);
  v += __shfl_xor(v, 4, 32);
  v += __shfl_xor(v, 8, 32);
  v += __shfl_xor(v, 16, 32);
  __shared__ float s[4];
  if ((h & 31) == 0) s[h >> 5] = v;
  __syncthreads();
  if (h == 0) tg[g] = s[0] + s[1] + s[2] + s[3];
}

// out[n] = sum_h W5[0][H+h]*relu( sum_k emb[n][k]*W7[h][k] ) + tg[batch[n]] + b5
__global__ __launch_bounds__(256) void final_gemm_kernel(
    const float* __restrict__ emb, const float* __restrict__ W7,
    const float* __restrict__ W5, const float* __restrict__ b5,
    const float* __restrict__ tg, const int* __restrict__ batch,
    float* __restrict__ out) {
  __shared__ v2f ldsB[8192];           // 64KB: W7 fragments, pre-swizzled
  stage_B(W7, ldsB);

  const int lane = threadIdx.x & 31;
  const int wave = threadIdx.x >> 5;
  const int tile = blockIdx.x * 8 + wave;
  if (tile >= NTILES) return;
  const int lane16 = lane & 15;
  const int hi = lane >> 4;
  const int row_base = tile * 16;

  v8f acc[8];
  #pragma unroll
  for (int j = 0; j < 8; j++)
    #pragma unroll
    for (int r = 0; r < 8; r++) acc[j][r] = 0.f;

  const float* aPtr = emb + (row_base + lane16) * HDIM + 2 * hi;
  const v2f* bLane = ldsB + lane;

  #pragma unroll 4
  for (int kk = 0; kk < 32; kk++) {
    v2f a = *(const v2f*)(aPtr + 4 * kk);
    #pragma unroll
    for (int j = 0; j < 8; j++) {
      v2f b = bLane[(kk * 8 + j) * 32];
      acc[j] = __builtin_amdgcn_wmma_f32_16x16x4_f32(
          false, a, false, b, (short)0, acc[j], false, false);
    }
  }

  float w5v[8];
  #pragma unroll
  for (int j = 0; j < 8; j++) w5v[j] = W5[HDIM + j * 16 + lane16];
  float bias = b5[0];

  #pragma unroll
  for (int r = 0; r < 8; r++) {
    float part = 0.f;
    #pragma unroll
    for (int j = 0; j < 8; j++) part += fmaxf(acc[j][r], 0.f) * w5v[j];
    // butterfly reduce across the 16 lanes of each half-wave
    part += __shfl_xor(part, 1, 32);
    part += __shfl_xor(part, 2, 32);
    part += __shfl_xor(part, 4, 32);
    part += __shfl_xor(part, 8, 32);
    int row = row_base + r + 8 * hi;
    if (lane16 == 0) out[row] = part + tg[batch[row]] + bias;
  }
}

extern "C" void kernel_launch(void* const* d_in, const int* in_sizes, int n_in,
                              void* d_out, int out_size, void* d_ws, size_t ws_size,
                              hipStream_t stream) {
  (void)in_sizes; (void)n_in; (void)out_size; (void)ws_size;
  const float* x     = (const float*)d_in[0];
  const int*   ei    = (const int*)d_in[1];
  const float* ea    = (const float*)d_in[2];
  const int*   batch = (const int*)d_in[3];
  const float* W1    = (const float*)d_in[4];
  const float* W2    = (const float*)d_in[5];
  const float* W3    = (const float*)d_in[6];
  const float* W4    = (const float*)d_in[7];
  const float* W5    = (const float*)d_in[8];
  const float* b5    = (const float*)d_in[9];
  const float* W6    = (const float*)d_in[10];
  const float* W7    = (const float*)d_in[11];
  float* out = (float*)d_out;

  float* ws      = (float*)d_ws;
  float* embA    = ws;                         // N*128
  float* embB    = embA + N_NODES * HDIM;      // N*128
  float* aggr    = embB + N_NODES * HDIM;      // N*128
  float* p       = aggr + N_NODES * HDIM;      // N
  float* m       = p + N_NODES;                // N
  float* pooled  = m + N_NODES;                // 16*128
  float* tg      = pooled + GRAPHS * HDIM;     // 16
  float* sum_emb = tg + GRAPHS;                // 16*128
  float* u3p     = sum_emb + GRAPHS * HDIM;    // 3*128
  float* u3m     = u3p + HOPS * HDIM;          // 3*128

  // zero p, m, pooled, tg (contiguous block)
  int nz = 2 * N_NODES + GRAPHS * HDIM + GRAPHS;
  zero_kernel<<<(nz + 255) / 256, 256, 0, stream>>>(p, nz);

  edge_scalar_kernel<<<(N_EDGES + 255) / 256, 256, 0, stream>>>(ei, ea, p, m);
  precompute_u_kernel<<<2, 256, 0, stream>>>(W3, W4, u3p, u3m);
  hop0_kernel<<<(N_NODES * HDIM + 255) / 256, 256, 0, stream>>>(x, W1, p, m, u3p, u3m, embA);

  float* cur = embA;
  float* nxt = embB;
  for (int l = 1; l < HOPS; l++) {
    zero_kernel<<<(N_NODES * HDIM + 255) / 256, 256, 0, stream>>>(aggr, N_NODES * HDIM);
    scatter_kernel<<<(N_EDGES * 32) / 256, 256, 0, stream>>>(ei, cur, aggr);
    gemm_hop_kernel<<<(NTILES + 7) / 8, 256, 0, stream>>>(
        aggr, W2 + l * HDIM * HDIM, x, W1 + l * HDIM, p, m,
        u3p + l * HDIM, u3m + l * HDIM, nxt);
    float* t = cur; cur = nxt; nxt = t;
  }

  pool_kernel<<<512, 256, 0, stream>>>(cur, batch, pooled);
  sumemb_kernel<<<(GRAPHS * HDIM + 255) / 256, 256, 0, stream>>>(pooled, W6, sum_emb);
  tg_kernel<<<GRAPHS, HDIM, 0, stream>>>(sum_emb, W5, tg);
  final_gemm_kernel<<<(NTILES + 7) / 8, 256, 0, stream>>>(cur, W7, W5, b5, tg, batch, out);
}